// ScaledDotProductAttention_32710470926661
// MI455X (gfx1250) — compile-verified
//
#include <hip/hip_runtime.h>
#include <math.h>

typedef __attribute__((ext_vector_type(2))) float v2f;
typedef __attribute__((ext_vector_type(4))) float v4f;
typedef __attribute__((ext_vector_type(8))) float v8f;

#define S_LEN    2048
#define D_HEAD   64
#define KTILES   128           // S/16
#define QK_SCALE 0.125f        // 1/sqrt(64)

__device__ __forceinline__ float row_max16(float v) {
#pragma unroll
  for (int m = 1; m < 16; m <<= 1) v = fmaxf(v, __shfl_xor(v, m, 32));
  return v;
}

#define WMMA_F32(a, b, c) \
  __builtin_amdgcn_wmma_f32_16x16x4_f32(false, (a), false, (b), (short)0, (c), false, false)

// One wave32 per 16-row query tile. blockDim = 32.
__global__ __launch_bounds__(32)
void attn_fwd_wmma_f32(const float* __restrict__ Q,
                       const float* __restrict__ K,
                       const float* __restrict__ V,
                       float* __restrict__ Out,
                       float* __restrict__ W)
{
  __shared__ float Ps[2 * 16 * 18];  // two P tiles, C-layout -> A-layout bounce

  const int lane = threadIdx.x;      // 0..31
  const int n    = lane & 15;        // N / column index
  const int h    = lane >> 4;        // half-wave id

  const int bid = blockIdx.x;
  const int qt  = bid & (KTILES - 1);
  const int bh  = bid >> 7;
  const int q0  = qt * 16;

  const float* Qb = Q + (size_t)bh * S_LEN * D_HEAD;
  const float* Kb = K + (size_t)bh * S_LEN * D_HEAD;
  const float* Vb = V + (size_t)bh * S_LEN * D_HEAD;
  float*       Ob = Out + (size_t)bh * S_LEN * D_HEAD;
  float*       Wb = W   + (size_t)bh * S_LEN * S_LEN;

  // ---- Q tile A-operands held in registers for the whole kernel ----
  // A layout: lane -> row M = n, K pair = 4*ks + 2h .. +1
  v2f qa[16];
  {
    const float* Qrow = Qb + (size_t)(q0 + n) * D_HEAD + 2 * h;
#pragma unroll
    for (int ks = 0; ks < 16; ++ks) qa[ks] = *(const v2f*)&Qrow[ks * 4];
  }

  // score tile: 16 x V_WMMA_F32_16X16X4_F32 over D=64, pre-scaled
  auto qk_tile = [&](int kt) -> v8f {
    v8f c = {};
    const float* Krow = Kb + (size_t)(kt * 16 + n) * D_HEAD + 2 * h;
#pragma unroll
    for (int ks = 0; ks < 16; ++ks) {
      v2f b = *(const v2f*)&Krow[ks * 4];   // B: N=n(key), K=4*ks+2h..+1
      c = WMMA_F32(qa[ks], b, c);
    }
#pragma unroll
    for (int r = 0; r < 8; ++r) c[r] *= QK_SCALE;
    return c;
  };
  auto mask_diag = [&](v8f& c) {            // causal mask on the diagonal tile
#pragma unroll
    for (int r = 0; r < 8; ++r)
      if (n > r + 8 * h) c[r] = -__builtin_inff();
  };

  v8f acc[5];  // acc[0..3]: O columns; acc[4]: row-sum l via ones-column WMMA
#pragma unroll
  for (int t = 0; t < 5; ++t) acc[t] = (v8f){0.f,0.f,0.f,0.f,0.f,0.f,0.f,0.f};
  float mrow[8];
#pragma unroll
  for (int r = 0; r < 8; ++r) mrow[r] = -__builtin_inff();

  // ============ Pass 1: flash-style, 2 k-tiles per stats/rescale round ============
  const int nchunk = (qt + 2) >> 1;
  for (int ch = 0; ch < nchunk; ++ch) {
    const int kt0 = 2 * ch, kt1 = 2 * ch + 1;

    v8f c0 = qk_tile(kt0);
    if (kt0 == qt) mask_diag(c0);
    v8f c1;
    if (kt1 <= qt) {
      c1 = qk_tile(kt1);
      if (kt1 == qt) mask_diag(c1);
    } else {
#pragma unroll
      for (int r = 0; r < 8; ++r) c1[r] = -__builtin_inff();  // phantom tile
    }

    // one max-reduction + one rescale per 32 keys
    float alpha[8];
#pragma unroll
    for (int r = 0; r < 8; ++r) {
      float mt = row_max16(fmaxf(c0[r], c1[r]));
      float mn = fmaxf(mrow[r], mt);
      alpha[r] = __expf(mrow[r] - mn);      // 0 on first chunk
      mrow[r]  = mn;
      c0[r]    = __expf(c0[r] - mn);        // masked/phantom -> 0
      c1[r]    = __expf(c1[r] - mn);
    }

    // bounce both P tiles through LDS (C-layout -> A-layout)
#pragma unroll
    for (int r = 0; r < 8; ++r) {
      Ps[(r + 8 * h) * 18 + n]       = c0[r];
      Ps[288 + (r + 8 * h) * 18 + n] = c1[r];
    }
    __syncthreads();

#pragma unroll
    for (int t = 0; t < 5; ++t)
#pragma unroll
      for (int r = 0; r < 8; ++r) acc[t][r] *= alpha[r];

#pragma unroll
    for (int half = 0; half < 2; ++half) {
      const int ktx = 2 * ch + half;
      if (ktx > qt) break;                  // skip phantom tile
#pragma unroll
      for (int ks = 0; ks < 4; ++ks) {
        v2f a = *(const v2f*)&Ps[half * 288 + n * 18 + ks * 4 + 2 * h];
        const float* Vk = Vb + (size_t)(ktx * 16 + ks * 4 + 2 * h) * D_HEAD;
#pragma unroll
        for (int t = 0; t < 4; ++t) {
          v2f b;
          b.x = Vk[t * 16 + n];
          b.y = Vk[D_HEAD + t * 16 + n];
          acc[t] = WMMA_F32(a, b, acc[t]);
        }
        v2f ones; ones.x = 1.f; ones.y = 1.f;
        acc[4] = WMMA_F32(a, ones, acc[4]); // row-sum l, no shuffles needed
      }
    }
    __syncthreads();
  }

  float linv[8];
#pragma unroll
  for (int r = 0; r < 8; ++r) linv[r] = 1.f / acc[4][r];

  // ============ Pass 2: recompute scores (L2-hot), stream normalized weights ======
  for (int kt = 0; kt <= qt; ++kt) {
    v8f c = {};
    const float* Krow = Kb + (size_t)(kt * 16 + n) * D_HEAD + 2 * h;
#pragma unroll
    for (int ks = 0; ks < 16; ++ks) {
      v2f b = *(const v2f*)&Krow[ks * 4];
      c = WMMA_F32(qa[ks], b, c);
    }
#pragma unroll
    for (int r = 0; r < 8; ++r) {
      float w = __expf(c[r] * QK_SCALE - mrow[r]) * linv[r];
      if (kt == qt && n > r + 8 * h) w = 0.f;
      Wb[(size_t)(q0 + r + 8 * h) * S_LEN + kt * 16 + n] = w;
    }
  }

  // ---- strictly-masked region: contiguous per-row span, fill with b128 stores ----
  {
    const int   cst = (qt + 1) * 16;          // first fully-masked column
    const int   L4  = (S_LEN - cst) >> 2;     // float4 slots per row
    const v4f   z   = (v4f){0.f, 0.f, 0.f, 0.f};
#pragma unroll
    for (int r = 0; r < 8; ++r) {
      float* row = Wb + (size_t)(q0 + r + 8 * h) * S_LEN + cst;
      for (int s = n; s < L4; s += 16)
        *(v4f*)&row[s * 4] = z;
    }
  }

  // ============ epilogue: normalized output ============
#pragma unroll
  for (int r = 0; r < 8; ++r) {
    const size_t row = (size_t)(q0 + r + 8 * h);
#pragma unroll
    for (int t = 0; t < 4; ++t)
      Ob[row * D_HEAD + t * 16 + n] = acc[t][r] * linv[r];
  }
}

extern "C" void kernel_launch(void* const* d_in, const int* in_sizes, int n_in,
                              void* d_out, int out_size, void* d_ws, size_t ws_size,
                              hipStream_t stream) {
  const float* Q = (const float*)d_in[0];
  const float* K = (const float*)d_in[1];
  const float* V = (const float*)d_in[2];
  // d_in[3] = causal mask; implemented analytically, not read.

  float* out = (float*)d_out;                    // [4,16,2048,64]
  float* wts = out + (size_t)4 * 16 * 2048 * 64; // [4,16,2048,2048]

  const int nblocks = 4 * 16 * KTILES;           // one wave32 per (b,h,q-tile)
  attn_fwd_wmma_f32<<<nblocks, 32, 0, stream>>>(Q, K, V, out, wts);
}